// TemporalEdgeSAGEClassifier_64312840290620
// MI455X (gfx1250) — compile-verified
//
#include <hip/hip_runtime.h>

#define N_NODES 50000
#define N_EDGES 300000
#define HDIM    256
#define EINDIM  64

typedef float v2f __attribute__((ext_vector_type(2)));
typedef float v8f __attribute__((ext_vector_type(8)));

__device__ __forceinline__ float sigmf(float x) { return 1.0f / (1.0f + expf(-x)); }

// ---------------------------------------------------------------------------
// fp32 WMMA GEMM:  C[M,N] = act(A[M,K] @ W[K,N] + bias)
// block = 256 threads (8 waves). Block tile: 64 rows x 128 cols.
// Each wave owns one 16-col N-tile and 4 M-subtiles (4 accumulators), so each
// B fragment (2 scalar loads, immediate offsets via template LDW) feeds 4
// V_WMMA_F32_16X16X4_F32. grid = (ceil(M/64), N/128), K multiple of 64.
// A-fragment ISA layout: lanes 0-15: M=lane, K=kb..kb+1; lanes 16-31:
// M=lane-16, K=kb+2..kb+3. B mirrors with N=lane&15.
// ---------------------------------------------------------------------------
template<int LDW>
__global__ __launch_bounds__(256)
void gemm_wmma_f32(const float* __restrict__ A, int lda, int M,
                   const float* __restrict__ W,
                   const float* __restrict__ bias,
                   float* __restrict__ C, int ldc,
                   int K, int act)
{
    __shared__ float As[64][68];   // pad 64 -> 68: conflict-free v2f frag reads
    const int tid   = threadIdx.x;
    const int wave  = tid >> 5;
    const int lane  = tid & 31;
    const int laneL = lane & 15;
    const int laneH = lane >> 4;
    const size_t m0 = (size_t)blockIdx.x * 64;
    const int nc    = blockIdx.y * 128 + wave * 16 + laneL;

    v8f acc[4];
    const v8f vz = {0.f,0.f,0.f,0.f,0.f,0.f,0.f,0.f};
    #pragma unroll
    for (int s = 0; s < 4; ++s) acc[s] = vz;

    for (int k0 = 0; k0 < K; k0 += 64) {
        __syncthreads();
        // stage A[m0:m0+64, k0:k0+64] as float4, zero-fill beyond M
        #pragma unroll
        for (int i = 0; i < 4; ++i) {
            int idx = tid + i * 256;            // 0..1023 float4 slots
            int r   = idx >> 4;                 // 16 float4 per row
            int c4  = (idx & 15) << 2;
            float4 v = make_float4(0.f, 0.f, 0.f, 0.f);
            if (m0 + r < (size_t)M)
                v = *(const float4*)&A[(m0 + r) * (size_t)lda + (k0 + c4)];
            As[r][c4 + 0] = v.x; As[r][c4 + 1] = v.y;
            As[r][c4 + 2] = v.z; As[r][c4 + 3] = v.w;
        }
        __syncthreads();
        // hoisted per-lane B base; all K-step offsets are immediates
        const float* wp = W + (size_t)(k0 + laneH * 2) * LDW + nc;
        #pragma unroll
        for (int kk = 0; kk < 64; kk += 4) {
            const int kb = kk + laneH * 2;
            v2f b; b.x = wp[kk * LDW]; b.y = wp[kk * LDW + LDW];
            #pragma unroll
            for (int s = 0; s < 4; ++s) {
                v2f a; a.x = As[s * 16 + laneL][kb]; a.y = As[s * 16 + laneL][kb + 1];
                acc[s] = __builtin_amdgcn_wmma_f32_16x16x4_f32(
                    false, a, false, b, (short)0, acc[s], false, false);
            }
        }
    }
    const float bv = bias ? bias[nc] : 0.0f;
    #pragma unroll
    for (int s = 0; s < 4; ++s) {
        const size_t mrow = m0 + s * 16 + laneH * 8;
        float* cp = C + mrow * (size_t)ldc + nc;
        #pragma unroll
        for (int r = 0; r < 8; ++r) {
            if (mrow + r < (size_t)M) {
                float v = acc[s][r] + bv;
                if (act) v = fmaxf(v, 0.0f);
                cp[(size_t)r * ldc] = v;
            }
        }
    }
}

// ---------------------------------------------------------------------------
// Degree count (scatter-mean denominator)
// ---------------------------------------------------------------------------
__global__ void degree_kernel(const int* __restrict__ dst, float* __restrict__ cnt, int E)
{
    int e = blockIdx.x * blockDim.x + threadIdx.x;
    if (e < E) atomicAdd(&cnt[dst[e]], 1.0f);
}

// ---------------------------------------------------------------------------
// Fused edge message: gate MLP (via precomputed node/edge linear parts),
// relu(msg)*gate, atomic scatter-add into agg[dst]. One wave per edge.
// msg_raw[e] = msgS[src] + msgD[dst] + msgE[e]
// gate pre   = gD[dst] + gS[src] + gE[e] + g1b  (gin = [h_dst, h_src, ea])
// ---------------------------------------------------------------------------
__global__ __launch_bounds__(256)
void edge_message_kernel(const int* __restrict__ src, const int* __restrict__ dst,
                         const float* __restrict__ msgS, const float* __restrict__ msgD,
                         const float* __restrict__ msgE,
                         const float* __restrict__ gS, const float* __restrict__ gD,
                         const float* __restrict__ gE,
                         const float* __restrict__ g1b, const float* __restrict__ g2w,
                         const float* __restrict__ g2b,
                         float* __restrict__ agg, int E)
{
    const int e    = (int)(((size_t)blockIdx.x * blockDim.x + threadIdx.x) >> 5);
    const int lane = threadIdx.x & 31;
    if (e >= E) return;
    const int s = src[e], d = dst[e];

    float gp = 0.0f;
    #pragma unroll
    for (int i = 0; i < 4; ++i) {
        int c = lane + 32 * i;
        float u = gD[(size_t)d * 128 + c] + gS[(size_t)s * 128 + c]
                + gE[(size_t)e * 128 + c] + g1b[c];
        u = fmaxf(u, 0.0f);
        gp += u * g2w[c];
    }
    for (int m = 16; m > 0; m >>= 1) gp += __shfl_xor(gp, m, 32);
    const float gate = sigmf(gp + g2b[0]);

    #pragma unroll
    for (int i = 0; i < 8; ++i) {
        int c = lane + 32 * i;
        float mv = msgS[(size_t)s * HDIM + c] + msgD[(size_t)d * HDIM + c]
                 + msgE[(size_t)e * HDIM + c];
        mv = fmaxf(mv, 0.0f) * gate;
        atomicAdd(&agg[(size_t)d * HDIM + c], mv);
    }
}

// ---------------------------------------------------------------------------
// Node update: out = relu(layernorm(agg/cnt + selfH)). One wave per row.
// ---------------------------------------------------------------------------
__global__ __launch_bounds__(256)
void node_update_kernel(const float* __restrict__ agg, const float* __restrict__ cnt,
                        const float* __restrict__ selfH,
                        const float* __restrict__ lng, const float* __restrict__ lnb,
                        float* __restrict__ hout, int N)
{
    const int row  = (int)(((size_t)blockIdx.x * blockDim.x + threadIdx.x) >> 5);
    const int lane = threadIdx.x & 31;
    if (row >= N) return;
    const float inv_c = 1.0f / fmaxf(cnt[row], 1.0f);
    const size_t base = (size_t)row * HDIM;

    float v[8];
    float s = 0.0f;
    #pragma unroll
    for (int i = 0; i < 8; ++i) {
        int c = lane + 32 * i;
        v[i] = agg[base + c] * inv_c + selfH[base + c];
        s += v[i];
    }
    for (int m = 16; m > 0; m >>= 1) s += __shfl_xor(s, m, 32);
    const float mu = s * (1.0f / HDIM);
    float var = 0.0f;
    #pragma unroll
    for (int i = 0; i < 8; ++i) { float d = v[i] - mu; var += d * d; }
    for (int m = 16; m > 0; m >>= 1) var += __shfl_xor(var, m, 32);
    const float inv = rsqrtf(var * (1.0f / HDIM) + 1e-5f);
    #pragma unroll
    for (int i = 0; i < 8; ++i) {
        int c = lane + 32 * i;
        float o = (v[i] - mu) * inv * lng[c] + lnb[c];
        hout[base + c] = fmaxf(o, 0.0f);
    }
}

// ---------------------------------------------------------------------------
// GRUCell elementwise (gi = h@Wih+bih, gh = hprev@Whh+bhh precomputed)
// ---------------------------------------------------------------------------
__global__ __launch_bounds__(256)
void gru_kernel(const float* __restrict__ gi, const float* __restrict__ gh,
                const float* __restrict__ hprev, float* __restrict__ ht, int N)
{
    size_t idx = (size_t)blockIdx.x * blockDim.x + threadIdx.x;
    if (idx >= (size_t)N * HDIM) return;
    size_t row = idx >> 8;
    int col = (int)(idx & 255);
    size_t b = row * 768;
    float r = sigmf(gi[b + col]       + gh[b + col]);
    float z = sigmf(gi[b + 256 + col] + gh[b + 256 + col]);
    float n = tanhf(gi[b + 512 + col] + r * gh[b + 512 + col]);
    ht[idx] = (1.0f - z) * n + z * hprev[idx];
}

// ---------------------------------------------------------------------------
// Fused edge classifier. One block (8 waves) per 16-edge tile.
// acc = nS[src] + nD[dst] + eE[e] + e1b  (precomputed linear parts of zf)
//     + |hs-hd| @ W3 + (hs*hd) @ W4      (per-edge WMMA, K=256, fully
//       unrolled so W3/W4 loads use immediate offsets)
// logits = relu(acc) @ e2w + e2b
// ---------------------------------------------------------------------------
__global__ __launch_bounds__(256)
void edge_classifier_kernel(const int* __restrict__ src, const int* __restrict__ dst,
                            const float* __restrict__ ht,
                            const float* __restrict__ nS, const float* __restrict__ nD,
                            const float* __restrict__ eE, const float* __restrict__ e1b,
                            const float* __restrict__ W3, const float* __restrict__ W4,
                            const float* __restrict__ e2w, const float* __restrict__ e2b,
                            float* __restrict__ logits, int E)
{
    __shared__ float dA[16][260];   // |hs-hd| tile (pad 4 mod 64 -> conflict-free)
    __shared__ float pA[16][260];   // hs*hd tile
    __shared__ float uS[16][260];   // relu(acc) staging
    __shared__ int   sI[16], dI[16];

    const int tid = threadIdx.x;
    const int e0  = blockIdx.x * 16;
    if (tid < 16) { sI[tid] = src[e0 + tid]; dI[tid] = dst[e0 + tid]; }
    __syncthreads();

    for (int i = tid; i < 16 * 256; i += 256) {
        int r = i >> 8, c = i & 255;
        float a = ht[(size_t)sI[r] * HDIM + c];
        float b = ht[(size_t)dI[r] * HDIM + c];
        dA[r][c] = fabsf(a - b);
        pA[r][c] = a * b;
    }
    __syncthreads();

    const int wave  = tid >> 5;
    const int lane  = tid & 31;
    const int laneL = lane & 15;
    const int laneH = lane >> 4;
    const int nb0   = wave * 32 + laneL;

    v8f acc0 = {0.f,0.f,0.f,0.f,0.f,0.f,0.f,0.f};
    v8f acc1 = acc0;
    #pragma unroll
    for (int r = 0; r < 8; ++r) {
        const int er = laneH * 8 + r;
        const int e  = e0 + er;
        const int s  = sI[er], d = dI[er];
        acc0[r] = nS[(size_t)s * HDIM + nb0] + nD[(size_t)d * HDIM + nb0]
                + eE[(size_t)e * HDIM + nb0] + e1b[nb0];
        acc1[r] = nS[(size_t)s * HDIM + nb0 + 16] + nD[(size_t)d * HDIM + nb0 + 16]
                + eE[(size_t)e * HDIM + nb0 + 16] + e1b[nb0 + 16];
    }

    // hoisted per-lane bases; full unroll makes every offset an immediate
    const float* w3p = W3 + (size_t)(laneH * 2) * HDIM + nb0;
    const float* w4p = W4 + (size_t)(laneH * 2) * HDIM + nb0;
    #pragma unroll
    for (int k = 0; k < 256; k += 4) {
        const int kb = k + laneH * 2;
        v2f ad; ad.x = dA[laneL][kb]; ad.y = dA[laneL][kb + 1];
        v2f ap; ap.x = pA[laneL][kb]; ap.y = pA[laneL][kb + 1];
        v2f b0; b0.x = w3p[k * HDIM];      b0.y = w3p[k * HDIM + HDIM];
        v2f b1; b1.x = w3p[k * HDIM + 16]; b1.y = w3p[k * HDIM + HDIM + 16];
        acc0 = __builtin_amdgcn_wmma_f32_16x16x4_f32(false, ad, false, b0, (short)0, acc0, false, false);
        acc1 = __builtin_amdgcn_wmma_f32_16x16x4_f32(false, ad, false, b1, (short)0, acc1, false, false);
        v2f c0; c0.x = w4p[k * HDIM];      c0.y = w4p[k * HDIM + HDIM];
        v2f c1; c1.x = w4p[k * HDIM + 16]; c1.y = w4p[k * HDIM + HDIM + 16];
        acc0 = __builtin_amdgcn_wmma_f32_16x16x4_f32(false, ap, false, c0, (short)0, acc0, false, false);
        acc1 = __builtin_amdgcn_wmma_f32_16x16x4_f32(false, ap, false, c1, (short)0, acc1, false, false);
    }

    #pragma unroll
    for (int r = 0; r < 8; ++r) {
        const int row = laneH * 8 + r;
        uS[row][wave * 32 + laneL]      = fmaxf(acc0[r], 0.0f);
        uS[row][wave * 32 + laneL + 16] = fmaxf(acc1[r], 0.0f);
    }
    __syncthreads();

    // 256 -> 2 head: each wave reduces 2 edges
    #pragma unroll
    for (int j = 0; j < 2; ++j) {
        const int er = wave * 2 + j;
        float p0 = 0.0f, p1 = 0.0f;
        #pragma unroll
        for (int i = 0; i < 8; ++i) {
            int c = lane + 32 * i;
            float u = uS[er][c];
            p0 += u * e2w[(size_t)c * 2];
            p1 += u * e2w[(size_t)c * 2 + 1];
        }
        for (int m = 16; m > 0; m >>= 1) {
            p0 += __shfl_xor(p0, m, 32);
            p1 += __shfl_xor(p1, m, 32);
        }
        if (lane == 0) {
            logits[(size_t)(e0 + er) * 2]     = p0 + e2b[0];
            logits[(size_t)(e0 + er) * 2 + 1] = p1 + e2b[1];
        }
    }
}

// ---------------------------------------------------------------------------
// Host side
// ---------------------------------------------------------------------------
static inline void gemm(hipStream_t s, const float* A, int lda, int M,
                        const float* W, int ldw, const float* bias,
                        float* C, int ldc, int N, int K, int act)
{
    dim3 g((M + 63) / 64, N / 128);
    if (ldw == 128)
        gemm_wmma_f32<128><<<g, 256, 0, s>>>(A, lda, M, W, bias, C, ldc, K, act);
    else if (ldw == 256)
        gemm_wmma_f32<256><<<g, 256, 0, s>>>(A, lda, M, W, bias, C, ldc, K, act);
    else
        gemm_wmma_f32<768><<<g, 256, 0, s>>>(A, lda, M, W, bias, C, ldc, K, act);
}

extern "C" void kernel_launch(void* const* d_in, const int* in_sizes, int n_in,
                              void* d_out, int out_size, void* d_ws, size_t ws_size,
                              hipStream_t stream)
{
    const int N = N_NODES, E = N_EDGES;

    const float* x         = (const float*)d_in[0];
    const int*   ei        = (const int*)d_in[1];
    const float* edge_attr = (const float*)d_in[2];
    const float* h_prev    = (const float*)d_in[3];
    const int* src = ei;          // edge_index[0]
    const int* dst = ei + E;      // edge_index[1]

    const float* L[2][11];
    for (int l = 0; l < 2; ++l)
        for (int i = 0; i < 11; ++i)
            L[l][i] = (const float*)d_in[4 + l * 11 + i];
    // per layer: 0 Wsrc, 1 Wdst, 2 Wedge, 3 Wself, 4 bself,
    //            5 g1w, 6 g1b, 7 g2w, 8 g2b, 9 lng, 10 lnb
    const float* gru_wih = (const float*)d_in[26];
    const float* gru_whh = (const float*)d_in[27];
    const float* gru_bih = (const float*)d_in[28];
    const float* gru_bhh = (const float*)d_in[29];
    const float* e1w     = (const float*)d_in[30];
    const float* e1b     = (const float*)d_in[31];
    const float* e2w     = (const float*)d_in[32];
    const float* e2b     = (const float*)d_in[33];

    float* out    = (float*)d_out;
    float* logits = out;                        // [E,2]
    float* h_t    = out + (size_t)E * 2;        // [N,256]

    // workspace carve-up
    float* ws = (float*)d_ws;
    size_t off = 0;
    auto take = [&](size_t n) { float* p = ws + off; off += (n + 255) & ~(size_t)255; return p; };
    float* cnt   = take(N);
    float* msgS  = take((size_t)N * HDIM);
    float* msgD  = take((size_t)N * HDIM);
    float* selfH = take((size_t)N * HDIM);
    float* gDb   = take((size_t)N * 128);
    float* gSb   = take((size_t)N * 128);
    float* agg   = take((size_t)N * HDIM);
    float* h1    = take((size_t)N * HDIM);
    float* h2    = take((size_t)N * HDIM);
    float* ebuf  = take((size_t)E * 384);       // reused region
    float* msgE = ebuf;                         // E x 256
    float* gE   = ebuf + (size_t)E * 256;       // E x 128
    float* gi   = ebuf;                         // N x 768 (after layers)
    float* gh   = ebuf + (size_t)N * 768;       // N x 768
    float* eEb  = ebuf;                         // E x 256 (classifier)
    (void)ws_size; (void)in_sizes; (void)n_in; (void)out_size;

    // degree (shared by both layers)
    hipMemsetAsync(cnt, 0, (size_t)N * sizeof(float), stream);
    degree_kernel<<<(E + 255) / 256, 256, 0, stream>>>(dst, cnt, E);

    const float* hin = x;
    float* houts[2] = { h1, h2 };
    for (int l = 0; l < 2; ++l) {
        const float** P = (const float**)L[l];
        // node-level linear parts (cin = 256 both layers)
        gemm(stream, hin, HDIM, N, P[0], HDIM, nullptr, msgS, HDIM, HDIM, HDIM, 0);  // h@Wsrc
        gemm(stream, hin, HDIM, N, P[1], HDIM, nullptr, msgD, HDIM, HDIM, HDIM, 0);  // h@Wdst
        gemm(stream, hin, HDIM, N, P[3], HDIM, P[4],    selfH, HDIM, HDIM, HDIM, 0); // h@Wself+b
        gemm(stream, hin, HDIM, N, P[5],             128, nullptr, gDb, 128, 128, HDIM, 0); // g1w rows 0:256 (dst)
        gemm(stream, hin, HDIM, N, P[5] + 256 * 128, 128, nullptr, gSb, 128, 128, HDIM, 0); // g1w rows 256:512 (src)
        // edge-attr linear parts
        gemm(stream, edge_attr, EINDIM, E, P[2], HDIM, nullptr, msgE, HDIM, HDIM, EINDIM, 0);
        gemm(stream, edge_attr, EINDIM, E, P[5] + 512 * 128, 128, nullptr, gE, 128, 128, EINDIM, 0);

        hipMemsetAsync(agg, 0, (size_t)N * HDIM * sizeof(float), stream);
        edge_message_kernel<<<(E * 32) / 256, 256, 0, stream>>>(
            src, dst, msgS, msgD, msgE, gSb, gDb, gE, P[6], P[7], P[8], agg, E);
        node_update_kernel<<<(N * 32 + 255) / 256, 256, 0, stream>>>(
            agg, cnt, selfH, P[9], P[10], houts[l], N);
        hin = houts[l];
    }

    // GRU
    gemm(stream, h2,     HDIM, N, gru_wih, 768, gru_bih, gi, 768, 768, HDIM, 0);
    gemm(stream, h_prev, HDIM, N, gru_whh, 768, gru_bhh, gh, 768, 768, HDIM, 0);
    gru_kernel<<<((size_t)N * HDIM + 255) / 256, 256, 0, stream>>>(gi, gh, h_prev, h_t, N);

    // edge classifier: precompute linear parts of zf = [hs, hd, ea, |hs-hd|, hs*hd]
    gemm(stream, h_t, HDIM, N, e1w,              HDIM, nullptr, msgS, HDIM, HDIM, HDIM, 0); // rows 0:256 (hs)
    gemm(stream, h_t, HDIM, N, e1w + 256 * HDIM, HDIM, nullptr, msgD, HDIM, HDIM, HDIM, 0); // rows 256:512 (hd)
    gemm(stream, edge_attr, EINDIM, E, e1w + 512 * HDIM, HDIM, nullptr, eEb, HDIM, HDIM, EINDIM, 0);

    edge_classifier_kernel<<<E / 16, 256, 0, stream>>>(
        src, dst, h_t, msgS, msgD, eEb, e1b,
        e1w + 576 * HDIM, e1w + 832 * HDIM, e2w, e2b, logits, E);
}